// FeatureWarpingLayer_32538672235128
// MI455X (gfx1250) — compile-verified
//
#include <hip/hip_runtime.h>

#define Bn 8
#define Cn 128
#define Hn 256
#define Wn 320
#define HWn (Hn * Wn)

typedef float v2f __attribute__((ext_vector_type(2)));
typedef float v8f __attribute__((ext_vector_type(8)));

__device__ __forceinline__ int iclamp(int v, int lo, int hi) {
    return v < lo ? lo : (v > hi ? hi : v);
}

// ---------------------------------------------------------------------------
// Per-batch matrix prep: ws[b*12 + 0..8] = M = K * R^T * K^{-1} (row major),
//                        ws[b*12 + 9..11] = Wv = K * R^T * (-t)
// ---------------------------------------------------------------------------
__global__ void warp_prep_kernel(const float* __restrict__ tvec,
                                 const float* __restrict__ Rm,
                                 const float* __restrict__ Km,
                                 float* __restrict__ ws) {
    int b = threadIdx.x;
    if (b >= Bn) return;
    const float* K = Km + b * 9;
    const float* R = Rm + b * 9;
    const float* t = tvec + b * 3;

    // 3x3 inverse of K via adjugate
    float a = K[0], bb = K[1], c = K[2];
    float d = K[3], e = K[4], f = K[5];
    float g = K[6], h = K[7], i = K[8];
    float A0 = e * i - f * h;
    float A1 = -(d * i - f * g);
    float A2 = d * h - e * g;
    float det = a * A0 + bb * A1 + c * A2;
    float inv = 1.0f / det;
    float Ki[9];
    Ki[0] = A0 * inv;              Ki[1] = -(bb * i - c * h) * inv; Ki[2] = (bb * f - c * e) * inv;
    Ki[3] = A1 * inv;              Ki[4] = (a * i - c * g) * inv;   Ki[5] = -(a * f - c * d) * inv;
    Ki[6] = A2 * inv;              Ki[7] = -(a * h - bb * g) * inv; Ki[8] = (a * e - bb * d) * inv;

    // T = K * R^T
    float T[9];
    for (int r = 0; r < 3; ++r)
        for (int cc = 0; cc < 3; ++cc)
            T[r * 3 + cc] = K[r * 3 + 0] * R[cc * 3 + 0] +
                            K[r * 3 + 1] * R[cc * 3 + 1] +
                            K[r * 3 + 2] * R[cc * 3 + 2];

    float* o = ws + b * 12;
    for (int r = 0; r < 3; ++r)
        for (int cc = 0; cc < 3; ++cc)
            o[r * 3 + cc] = T[r * 3 + 0] * Ki[0 * 3 + cc] +
                            T[r * 3 + 1] * Ki[1 * 3 + cc] +
                            T[r * 3 + 2] * Ki[2 * 3 + cc];
    for (int r = 0; r < 3; ++r)
        o[9 + r] = -(T[r * 3 + 0] * t[0] + T[r * 3 + 1] * t[1] + T[r * 3 + 2] * t[2]);
}

// ---------------------------------------------------------------------------
// Main warp kernel. One wave32 = 16 pixels of one row. The projective
// transform inter = M * [x,y,1] for the 16 pixels is done with a single
// V_WMMA_F32_16X16X4_F32: M occupies A rows {0,1,2} AND {8,9,10} so both
// lane halves receive pixel (lane&15)'s result in d[0..2] without shuffles.
// Lane j handles channels 0..63, lane j+16 handles channels 64..127.
// ---------------------------------------------------------------------------
__global__ __launch_bounds__(256) void warp_main_kernel(
    const float* __restrict__ depth, const float* __restrict__ feat,
    const float* __restrict__ mask, const float* __restrict__ ws,
    float* __restrict__ out_warp, float* __restrict__ out_inter) {

    const int lane   = threadIdx.x & 31;
    const int waveId = blockIdx.x * (blockDim.x >> 5) + (threadIdx.x >> 5);
    const int gw = waveId % (Wn / 16);
    const int h  = (waveId / (Wn / 16)) % Hn;
    const int b  = waveId / ((Wn / 16) * Hn);
    const int wbase = gw * 16;
    const int r16   = lane & 15;          // pixel within group / row-in-A
    const int x     = wbase + r16;        // this lane's pixel x

    const float* mw = ws + b * 12;        // M (9) + Wv (3)

    float iX, iY, iZ;
#if __has_builtin(__builtin_amdgcn_wmma_f32_16x16x4_f32)
    // A fragment (16x4): lane L<16 holds A[L][0],A[L][1]; lane L+16 holds A[L][2],A[L][3]
    const int  r      = r16 & 7;
    const bool hasRow = (r < 3) && ((r16 & 7) == r16 || r16 >= 8); // rows 0,1,2,8,9,10
    v2f afrag;
    if (lane < 16) {
        afrag.x = hasRow ? mw[r * 3 + 0] : 0.0f;
        afrag.y = hasRow ? mw[r * 3 + 1] : 0.0f;
    } else {
        afrag.x = hasRow ? mw[r * 3 + 2] : 0.0f;
        afrag.y = 0.0f;
    }
    // B fragment (4x16): lane j<16 holds B[0][j],B[1][j]; lane j+16 holds B[2][j],B[3][j]
    v2f bfrag;
    if (lane < 16) {
        bfrag.x = (float)x;               // grid x of column j
        bfrag.y = (float)h;               // grid y
    } else {
        bfrag.x = 1.0f;                   // homogeneous 1
        bfrag.y = 0.0f;                   // K=3 pad
    }
    v8f acc = {};
    acc = __builtin_amdgcn_wmma_f32_16x16x4_f32(false, afrag, false, bfrag,
                                                (short)0, acc, false, false);
    iX = acc[0];
    iY = acc[1];
    iZ = acc[2];
#else
    iX = mw[0] * (float)x + mw[1] * (float)h + mw[2];
    iY = mw[3] * (float)x + mw[4] * (float)h + mw[5];
    iZ = mw[6] * (float)x + mw[7] * (float)h + mw[8];
#endif

    const int dIdx = (b * Hn + h) * Wn + x;
    const float dep = depth[dIdx];
    const float m   = mask[dIdx];

    float d2 = mw[11] + dep * iZ;
    d2 = 1e30f * (1.0f - m) + m * d2;
    const float u2 = (mw[9]  + dep * iX) / d2;
    const float v2 = (mw[10] + dep * iY) / d2;

    // grid_sample(bilinear, zeros, align_corners=False): sample at (u-0.5, v-0.5)
    const float pxf = u2 - 0.5f, pyf = v2 - 0.5f;
    const float x0f = floorf(pxf), y0f = floorf(pyf);
    const float fx = pxf - x0f, fy = pyf - y0f;
    const int x0 = (int)x0f, y0 = (int)y0f;
    const int x1 = x0 + 1,   y1 = y0 + 1;

    const float vx0 = (x0 >= 0 && x0 < Wn) ? 1.0f : 0.0f;
    const float vx1 = (x1 >= 0 && x1 < Wn) ? 1.0f : 0.0f;
    const float vy0 = (y0 >= 0 && y0 < Hn) ? 1.0f : 0.0f;
    const float vy1 = (y1 >= 0 && y1 < Hn) ? 1.0f : 0.0f;

    const int xc0 = iclamp(x0, 0, Wn - 1), xc1 = iclamp(x1, 0, Wn - 1);
    const int yc0 = iclamp(y0, 0, Hn - 1), yc1 = iclamp(y1, 0, Hn - 1);
    const int o00 = yc0 * Wn + xc0, o01 = yc0 * Wn + xc1;
    const int o10 = yc1 * Wn + xc0, o11 = yc1 * Wn + xc1;

    // validity folded into weights -> branch-free, always-in-bounds loads
    const float w00 = (1.0f - fx) * (1.0f - fy) * vx0 * vy0;
    const float w01 = fx * (1.0f - fy) * vx1 * vy0;
    const float w10 = (1.0f - fx) * fy * vx0 * vy1;
    const float w11 = fx * fy * vx1 * vy1;

    // intersection mask (bilinear-sample mask, threshold 0.9)
    const float* mb = mask + (size_t)b * HWn;
    const float ms = w00 * mb[o00] + w01 * mb[o01] + w10 * mb[o10] + w11 * mb[o11];
    const float inter = (ms * m >= 0.9f) ? 1.0f : 0.0f;
    if (lane < 16) __builtin_nontemporal_store(inter, &out_inter[dIdx]);

    // channel loop: lane j -> c in [0,64), lane j+16 -> c in [64,128)
    const int c0 = (lane < 16) ? 0 : (Cn / 2);
    const float* fp = feat + ((size_t)b * Cn + c0) * HWn;
    float* op = out_warp + ((size_t)b * Cn + c0) * HWn + (size_t)(h * Wn + x);
#pragma unroll 4
    for (int c = 0; c < Cn / 2; ++c) {
        const float v00 = fp[o00];
        const float v01 = fp[o01];
        const float v10 = fp[o10];
        const float v11 = fp[o11];
        const float r0 = w00 * v00 + w01 * v01 + w10 * v10 + w11 * v11;
        __builtin_nontemporal_store(r0, op);   // stream out, keep L2 for gathers
        fp += HWn;
        op += HWn;
    }
}

// ---------------------------------------------------------------------------
// Inputs: 0 depth [8,1,256,320], 1 feat [8,128,256,320], 2 masks [8,1,256,320],
//         3 t [8,3,1], 4 R [8,3,3], 5 K [8,3,3]
// Output: warped [8,128,256,320] ++ intersect [8,1,256,320]  (float32)
// ---------------------------------------------------------------------------
extern "C" void kernel_launch(void* const* d_in, const int* in_sizes, int n_in,
                              void* d_out, int out_size, void* d_ws, size_t ws_size,
                              hipStream_t stream) {
    const float* depth = (const float*)d_in[0];
    const float* feat  = (const float*)d_in[1];
    const float* mask  = (const float*)d_in[2];
    const float* tvec  = (const float*)d_in[3];
    const float* Rm    = (const float*)d_in[4];
    const float* Km    = (const float*)d_in[5];

    float* out_warp  = (float*)d_out;
    float* out_inter = out_warp + (size_t)Bn * Cn * HWn;
    float* wsf       = (float*)d_ws;   // 8 * 12 floats

    warp_prep_kernel<<<1, 32, 0, stream>>>(tvec, Rm, Km, wsf);

    const int totalWaves = Bn * Hn * (Wn / 16);  // 40960
    const int blocks     = totalWaves / 8;        // 8 waves per 256-thread block
    warp_main_kernel<<<blocks, 256, 0, stream>>>(depth, feat, mask, wsf,
                                                 out_warp, out_inter);
}